// OrthogonalLayer1D_64510408786204
// MI455X (gfx1250) — compile-verified
//
#include <hip/hip_runtime.h>

typedef __attribute__((ext_vector_type(2))) float v2f;
typedef __attribute__((ext_vector_type(8))) float v8f;

#define N_MODELS   4
#define N_SAMPLES  32768
#define DIM        256

#define WAVES_PER_WG     8
#define THREADS_PER_WG   (WAVES_PER_WG * 32)
#define SAMPLES_PER_WAVE 4
#define SAMPLES_PER_WG   (WAVES_PER_WG * SAMPLES_PER_WAVE)

// ---- LDS layout (in floats) ----
// B tile per wave: 16 rows x 260 (stride pad: 1040B rows -> 16B aligned, banks spread by 4)
#define BSTR     260
#define BT_WAVE  (16 * BSTR)                       // 4160
#define B_OFF    0
#define G_OFF    (WAVES_PER_WG * BT_WAVE)          // 33280
#define GSTR     17
#define G_WAVE   (16 * GSTR)                       // 272
#define Z_OFF    (G_OFF + WAVES_PER_WG * G_WAVE)   // 35456
#define ZSTR     18                                // even -> b64-aligned operand loads
#define Z_WAVE   (16 * ZSTR)                       // 288
#define LDS_FLOATS (Z_OFF + WAVES_PER_WG * Z_WAVE) // 37760 floats = 151040 B (< 160KB)

__device__ __forceinline__ void mat4_mul(float* d, const float* a, const float* b) {
#pragma unroll
  for (int i = 0; i < 4; ++i) {
#pragma unroll
    for (int j = 0; j < 4; ++j) {
      float s = a[i * 4 + 0] * b[0 + j];
      s = fmaf(a[i * 4 + 1], b[4 + j], s);
      s = fmaf(a[i * 4 + 2], b[8 + j], s);
      s = fmaf(a[i * 4 + 3], b[12 + j], s);
      d[i * 4 + j] = s;
    }
  }
}

__global__ __launch_bounds__(THREADS_PER_WG)
void orth_polar_kernel(const float* __restrict__ x, float* __restrict__ out) {
  __shared__ float lds[LDS_FLOATS];

  const int tid  = threadIdx.x;
  const int wave = tid >> 5;
  const int lane = tid & 31;
  const int Lr   = lane & 15;   // row/col index within a 16-half
  const int h    = lane >> 4;   // 0: lanes 0-15, 1: lanes 16-31
  const int s0   = blockIdx.x * SAMPLES_PER_WG + wave * SAMPLES_PER_WAVE;

  float* bt = lds + B_OFF + wave * BT_WAVE;

  // ---------- Phase 1: global -> LDS, 16x256 tile (rows: sample-major, 4 models) ----------
#pragma unroll
  for (int r = 0; r < 16; ++r) {
    const int    smp   = s0 + (r >> 2);
    const int    mdl   = r & 3;
    const size_t gbase = (size_t)mdl * (size_t)N_SAMPLES * DIM + (size_t)smp * DIM;
#pragma unroll
    for (int half = 0; half < 2; ++half) {
      const int c = half * 128 + lane * 4;            // fully coalesced 512B per wave-instr
      float4 v = *(const float4*)(x + gbase + c);
      *(float4*)(bt + r * BSTR + c) = v;              // 16B-aligned LDS store
    }
  }
  // B tile is wave-local and same-wave LDS ops are in-order: no barrier needed here.

  // ---------- Zero-fill block-diagonal Z16 buffers (all threads, before barrier 1) ----------
  for (int i = tid; i < WAVES_PER_WG * Z_WAVE; i += THREADS_PER_WG)
    lds[Z_OFF + i] = 0.0f;

  // ---------- Phase 2: G16 = B * B^T via V_WMMA_F32_16X16X4_F32 ----------
  // A-operand (16x4) lane L: { B[L%16, k+2h], B[L%16, k+2h+1] }
  // B-operand (4x16)  lane n: { B[n%16, k+2h], B[n%16, k+2h+1] }  -> identical registers!
  v8f acc = {};
  const float* arow = bt + Lr * BSTR + 2 * h;
#pragma unroll 8
  for (int k = 0; k < DIM; k += 4) {
    v2f ab = *(const v2f*)(arow + k);                 // one ds_load_b64 feeds both operands
    acc = __builtin_amdgcn_wmma_f32_16x16x4_f32(false, ab, false, ab,
                                                (short)0, acc, false, false);
  }

  // Store G16 (C layout: vgpr i -> row m = i + 8h, col n = lane%16)
  float* gw = lds + G_OFF + wave * G_WAVE;
#pragma unroll
  for (int i = 0; i < 8; ++i) gw[(i + 8 * h) * GSTR + Lr] = acc[i];

  __syncthreads();

  // ---------- Phase 3: Newton-Schulz G^{-1/2} (one lane per 4x4 block, 32 blocks/WG) ----------
  if (tid < 32) {
    const int wsrc = tid >> 2;                        // which wave's G area
    const int lb   = tid & 3;                         // block within that wave
    const float* gs = lds + G_OFF + wsrc * G_WAVE;

    float g[16];
#pragma unroll
    for (int i = 0; i < 4; ++i)
#pragma unroll
      for (int j = 0; j < 4; ++j)
        g[i * 4 + j] = gs[(4 * lb + i) * GSTR + (4 * lb + j)];

    const float tr = g[0] + g[5] + g[10] + g[15];     // = ||A||_F^2
    float fro2 = 0.0f;
#pragma unroll
    for (int e = 0; e < 16; ++e) fro2 = fmaf(g[e], g[e], fro2);

    float w[16];
    if (tr < 1e-16f) {                                // reference EPS passthrough: W = I
#pragma unroll
      for (int e = 0; e < 16; ++e) w[e] = (e % 5 == 0) ? 1.0f : 0.0f;
    } else {
      const float invc = rsqrtf(fro2);                // 1/||G||_F ; exact value cancels below
      float y[16], z[16], yn[16], zn[16], t[16];
#pragma unroll
      for (int e = 0; e < 16; ++e) {
        y[e] = g[e] * invc;
        z[e] = (e % 5 == 0) ? 1.0f : 0.0f;
      }
#pragma unroll 1
      for (int it = 0; it < 10; ++it) {               // quadratic convergence, >=fp32 eps
        mat4_mul(t, z, y);                            // t = Z*Y
#pragma unroll
        for (int e = 0; e < 16; ++e)
          t[e] = ((e % 5 == 0) ? 1.5f : 0.0f) - 0.5f * t[e];  // t = (3I - ZY)/2
        mat4_mul(yn, y, t);
        mat4_mul(zn, t, z);
#pragma unroll
        for (int e = 0; e < 16; ++e) { y[e] = yn[e]; z[e] = zn[e]; }
      }
      const float sc = sqrtf(invc);                   // Z_inf = (1/sqrt(invc)) * G^{-1/2}
#pragma unroll
      for (int e = 0; e < 16; ++e) w[e] = z[e] * sc;  // W = G^{-1/2}
    }

    float* zw = lds + Z_OFF + wsrc * Z_WAVE;          // scatter into block-diagonal Z16
#pragma unroll
    for (int i = 0; i < 4; ++i)
#pragma unroll
      for (int j = 0; j < 4; ++j)
        zw[(4 * lb + i) * ZSTR + (4 * lb + j)] = w[i * 4 + j];
  }

  __syncthreads();

  // ---------- Phase 4: out = Z16_blockdiag * B via WMMA, 16 column tiles ----------
  const float* zq = lds + Z_OFF + wave * Z_WAVE + Lr * ZSTR + 2 * h;
  for (int tc = 0; tc < 16; ++tc) {
    v8f c = {};
#pragma unroll
    for (int kc = 0; kc < 4; ++kc) {
      v2f a = *(const v2f*)(zq + 4 * kc);             // A-op: Z16[Lr, 4kc+2h .. +1]
      v2f b;                                          // B-op: B[4kc + j + 2h, tc*16 + Lr]
      b.x = bt[(4 * kc + 2 * h) * BSTR + tc * 16 + Lr];
      b.y = bt[(4 * kc + 2 * h + 1) * BSTR + tc * 16 + Lr];
      c = __builtin_amdgcn_wmma_f32_16x16x4_f32(false, a, false, b,
                                                (short)0, c, false, false);
    }
#pragma unroll
    for (int i = 0; i < 8; ++i) {                     // C layout: row m = i + 8h, col = lane%16
      const int m   = i + 8 * h;
      const int smp = s0 + (m >> 2);
      const int mdl = m & 3;
      out[(size_t)mdl * (size_t)N_SAMPLES * DIM + (size_t)smp * DIM + tc * 16 + Lr] = c[i];
    }
  }
}

extern "C" void kernel_launch(void* const* d_in, const int* in_sizes, int n_in,
                              void* d_out, int out_size, void* d_ws, size_t ws_size,
                              hipStream_t stream) {
  (void)in_sizes; (void)n_in; (void)out_size; (void)d_ws; (void)ws_size;
  const float* x  = (const float*)d_in[0];
  float*       out = (float*)d_out;
  dim3 grid(N_SAMPLES / SAMPLES_PER_WG);   // 1024 workgroups
  dim3 block(THREADS_PER_WG);              // 256 threads = 8 waves (wave32)
  orth_polar_kernel<<<grid, block, 0, stream>>>(x, out);
}